// GCN_22660247453949
// MI455X (gfx1250) — compile-verified
//
#include <hip/hip_runtime.h>
#include <math.h>

#define NNODES 100000
#define NEDGES 1600000
#define NGRAPHS 64
#define DIN 256
#define DHID 128
#define DOUT 64
#define EPSF 1e-5f

typedef __attribute__((ext_vector_type(2))) float v2f;
typedef __attribute__((ext_vector_type(8))) float v8f;

// ---------------------------------------------------------------- utilities
__global__ void zero_f32(float* __restrict__ p, int n) {
  int i = blockIdx.x * blockDim.x + threadIdx.x;
  if (i < n) p[i] = 0.0f;
}

__global__ void deg_accum(const int* __restrict__ col, const float* __restrict__ w,
                          float* __restrict__ deg, int nE) {
  int e = blockIdx.x * blockDim.x + threadIdx.x;
  if (e < nE) atomicAdd(&deg[col[e]], w[e]);
}

__global__ void deg_to_dinv(float* __restrict__ deg, int n) {
  int i = blockIdx.x * blockDim.x + threadIdx.x;
  if (i < n) {
    float d = deg[i] + 1.0f;                 // + self-loop weight 1
    deg[i] = rsqrtf(fmaxf(d, EPSF));         // deg >= 1 > 0 always
  }
}

// ---------------------------------------------------------------- WMMA GEMM
// C[M x NCOLS] = A[M x KDIM] * B[KDIM x NCOLS], fp32.
// One wave per 16-row stripe computing all NCOLS columns (A read once).
// B is cooperatively staged in LDS, pre-packed so that one lane fragment
// {B[k][n], B[k+1][n]} is a single contiguous float2 -> one ds_load_b64.
//
// V_WMMA_F32_16X16X4_F32 layouts (wave32):
//   A 16x4 : lane l(0..15): M=l, v0=K0,v1=K1 ; lane 16+l: M=l, v0=K2,v1=K3
//   B 4x16 : v0: lanes0-15 K=0 N=l, lanes16-31 K=2 ; v1: K=1 / K=3
//   C/D    : vgpr r: lanes0-15 -> (M=r, N=l), lanes16-31 -> (M=r+8, N=l)
template <int NCOLS, int KDIM>
__global__ void wmma_gemm_f32(const float* __restrict__ A, const float* __restrict__ B,
                              float* __restrict__ C, int M) {
  __shared__ float2 Bl[(KDIM / 2) * NCOLS];

  // cooperative pack: Bl[k2*NCOLS + n] = {B[2k2][n], B[2k2+1][n]}
  for (int i = threadIdx.x; i < (KDIM / 2) * NCOLS; i += blockDim.x) {
    const int k2 = i / NCOLS;
    const int n = i - k2 * NCOLS;
    float2 v;
    v.x = B[(size_t)(2 * k2 + 0) * NCOLS + n];
    v.y = B[(size_t)(2 * k2 + 1) * NCOLS + n];
    Bl[i] = v;
  }
  __syncthreads();

  const int wave = (int)(((size_t)blockIdx.x * blockDim.x + threadIdx.x) >> 5);
  const int lane = threadIdx.x & 31;
  if (wave >= M / 16) return;                 // wave-uniform exit: EXEC stays all-1s
  const int m0 = wave * 16;
  const int half = lane >> 4;
  const int l = lane & 15;
  constexpr int NT = NCOLS / 16;

  v8f acc[NT];
#pragma unroll
  for (int t = 0; t < NT; ++t)
#pragma unroll
    for (int j = 0; j < 8; ++j) acc[t][j] = 0.0f;

  const float* __restrict__ arow = A + (size_t)(m0 + l) * KDIM + 2 * half;

  // software-pipelined A fragment (one global_load_b64 per k-step)
  v2f a;
  a[0] = arow[0];
  a[1] = arow[1];

  for (int k0 = 0; k0 < KDIM; k0 += 4) {
    v2f a_next = a;
    if (k0 + 4 < KDIM) {
      a_next[0] = arow[k0 + 4 + 0];
      a_next[1] = arow[k0 + 4 + 1];
    }
    // batch all NT B-fragment LDS loads into independent registers
    const float2* __restrict__ brow = Bl + (size_t)(k0 / 2 + half) * NCOLS + l;
    v2f bfrag[NT];
#pragma unroll
    for (int t = 0; t < NT; ++t) {
      float2 bb = brow[t * 16];
      bfrag[t][0] = bb.x;
      bfrag[t][1] = bb.y;
    }
    // back-to-back WMMA burst
#pragma unroll
    for (int t = 0; t < NT; ++t)
      acc[t] = __builtin_amdgcn_wmma_f32_16x16x4_f32(
          /*neg_a=*/false, a, /*neg_b=*/false, bfrag[t],
          /*c_mod=*/(short)0, acc[t], /*reuse_a=*/false, /*reuse_b=*/false);
    a = a_next;
  }

#pragma unroll
  for (int t = 0; t < NT; ++t)
#pragma unroll
    for (int r = 0; r < 8; ++r)
      C[(size_t)(m0 + r + 8 * half) * NCOLS + t * 16 + l] = acc[t][r];
}

// ------------------------------------------------- self-loop + bias init
// out[i,:] = bias + h[i,:] * dinv[i]^2      (self-loop norm = dinv*1*dinv)
template <int D>
__global__ void init_out(const float* __restrict__ h, const float* __restrict__ dinv,
                         const float* __restrict__ bias, float* __restrict__ out, int n) {
  const int wave = (int)(((size_t)blockIdx.x * blockDim.x + threadIdx.x) >> 5);
  const int lane = threadIdx.x & 31;
  if (wave >= n) return;
  constexpr int F = D / 32;
  const float s = dinv[wave] * dinv[wave];
  const float* __restrict__ hp = h + (size_t)wave * D + lane * F;
  float* __restrict__ op = out + (size_t)wave * D + lane * F;
#pragma unroll
  for (int f = 0; f < F; ++f) op[f] = bias[lane * F + f] + hp[f] * s;
}

// ------------------------------------------------- edge scatter (wave/edge)
template <int D>
__global__ void edge_scatter(const float* __restrict__ h, const int* __restrict__ rows,
                             const int* __restrict__ cols, const float* __restrict__ w,
                             const float* __restrict__ dinv, float* __restrict__ out, int nE) {
  const int wave = (int)(((size_t)blockIdx.x * blockDim.x + threadIdx.x) >> 5);
  const int lane = threadIdx.x & 31;
  if (wave >= nE) return;
  const int r = rows[wave];
  const int c = cols[wave];
  const float nrm = dinv[r] * w[wave] * dinv[c];
  constexpr int F = D / 32;
  const float* __restrict__ hp = h + (size_t)r * D + lane * F;
  float* __restrict__ op = out + (size_t)c * D + lane * F;
#pragma unroll
  for (int f = 0; f < F; ++f) atomicAdd(op + f, hp[f] * nrm);
}

// ------------------------------------------------- relu + layernorm (in place)
template <int D>
__global__ void relu_layernorm(float* __restrict__ x, const float* __restrict__ g,
                               const float* __restrict__ b, int n) {
  const int wave = (int)(((size_t)blockIdx.x * blockDim.x + threadIdx.x) >> 5);
  const int lane = threadIdx.x & 31;
  if (wave >= n) return;
  constexpr int F = D / 32;
  float* __restrict__ p = x + (size_t)wave * D + lane * F;
  float v[F];
  float s = 0.0f;
#pragma unroll
  for (int f = 0; f < F; ++f) {
    v[f] = fmaxf(p[f], 0.0f);
    s += v[f];
  }
#pragma unroll
  for (int off = 16; off > 0; off >>= 1) s += __shfl_xor(s, off, 32);
  const float mu = s * (1.0f / D);
  float q = 0.0f;
#pragma unroll
  for (int f = 0; f < F; ++f) {
    float d = v[f] - mu;
    q += d * d;
  }
#pragma unroll
  for (int off = 16; off > 0; off >>= 1) q += __shfl_xor(q, off, 32);
  const float rs = rsqrtf(q * (1.0f / D) + EPSF);
#pragma unroll
  for (int f = 0; f < F; ++f)
    p[f] = (v[f] - mu) * rs * g[lane * F + f] + b[lane * F + f];
}

// ------------------------------------------------- global mean pool + sigmoid
__global__ void pool_accum(const float* __restrict__ a, const int* __restrict__ batch,
                           float* __restrict__ sums, float* __restrict__ cnts, int n) {
  const int wave = (int)(((size_t)blockIdx.x * blockDim.x + threadIdx.x) >> 5);
  const int lane = threadIdx.x & 31;
  if (wave >= n) return;
  const int g = batch[wave];
  const float* __restrict__ p = a + (size_t)wave * DOUT + lane * 2;
  atomicAdd(&sums[g * DOUT + lane * 2 + 0], p[0]);
  atomicAdd(&sums[g * DOUT + lane * 2 + 1], p[1]);
  if (lane == 0) atomicAdd(&cnts[g], 1.0f);
}

__global__ void pool_final(const float* __restrict__ sums, const float* __restrict__ cnts,
                           float* __restrict__ out) {
  int i = blockIdx.x * blockDim.x + threadIdx.x;
  if (i < NGRAPHS * DOUT) {
    int g = i / DOUT;
    float m = sums[i] / fmaxf(cnts[g], 1.0f);
    out[i] = 1.0f / (1.0f + __expf(-m));
  }
}

// ---------------------------------------------------------------- launcher
extern "C" void kernel_launch(void* const* d_in, const int* in_sizes, int n_in,
                              void* d_out, int out_size, void* d_ws, size_t ws_size,
                              hipStream_t stream) {
  const float* x   = (const float*)d_in[0];
  const int*   ei  = (const int*)d_in[1];
  const float* ew  = (const float*)d_in[2];
  const int*   bat = (const int*)d_in[3];
  const float* W1  = (const float*)d_in[4];
  const float* b1  = (const float*)d_in[5];
  const float* g1  = (const float*)d_in[6];
  const float* bt1 = (const float*)d_in[7];
  const float* W2  = (const float*)d_in[8];
  const float* b2  = (const float*)d_in[9];
  const float* g2  = (const float*)d_in[10];
  const float* bt2 = (const float*)d_in[11];
  float* out = (float*)d_out;

  const int* rows = ei;             // edge_index[0] = source
  const int* cols = ei + NEDGES;    // edge_index[1] = target

  // workspace carve-out (h1/h2 alias, a1/a2 alias)
  char* ws = (char*)d_ws;
  size_t off = 0;
  auto carve = [&](size_t bytes) -> float* {
    float* p = (float*)(ws + off);
    off += (bytes + 255) & ~(size_t)255;
    return p;
  };
  float* dinv = carve((size_t)NNODES * 4);
  float* h    = carve((size_t)NNODES * DHID * 4);   // h1, then h2
  float* a    = carve((size_t)NNODES * DHID * 4);   // a1, then a2
  float* sums = carve((size_t)NGRAPHS * DOUT * 4);
  float* cnts = carve((size_t)NGRAPHS * 4);

  const int TB = 256;
  auto blocks = [&](long long threads) { return (int)((threads + TB - 1) / TB); };

  // symmetric degree normalization (shared by both layers)
  zero_f32<<<blocks(NNODES), TB, 0, stream>>>(dinv, NNODES);
  deg_accum<<<blocks(NEDGES), TB, 0, stream>>>(cols, ew, dinv, NEDGES);
  deg_to_dinv<<<blocks(NNODES), TB, 0, stream>>>(dinv, NNODES);

  // ---- layer 1: GCNConv(256->128) -> relu -> LN
  wmma_gemm_f32<DHID, DIN><<<blocks((long long)(NNODES / 16) * 32), TB, 0, stream>>>(x, W1, h, NNODES);
  init_out<DHID><<<blocks((long long)NNODES * 32), TB, 0, stream>>>(h, dinv, b1, a, NNODES);
  edge_scatter<DHID><<<blocks((long long)NEDGES * 32), TB, 0, stream>>>(h, rows, cols, ew, dinv, a, NEDGES);
  relu_layernorm<DHID><<<blocks((long long)NNODES * 32), TB, 0, stream>>>(a, g1, bt1, NNODES);

  // ---- layer 2: GCNConv(128->64) -> relu -> LN
  wmma_gemm_f32<DOUT, DHID><<<blocks((long long)(NNODES / 16) * 32), TB, 0, stream>>>(a, W2, h, NNODES);
  init_out<DOUT><<<blocks((long long)NNODES * 32), TB, 0, stream>>>(h, dinv, b2, a, NNODES);
  edge_scatter<DOUT><<<blocks((long long)NEDGES * 32), TB, 0, stream>>>(h, rows, cols, ew, dinv, a, NEDGES);
  relu_layernorm<DOUT><<<blocks((long long)NNODES * 32), TB, 0, stream>>>(a, g2, bt2, NNODES);

  // ---- global mean pool + sigmoid
  zero_f32<<<blocks(NGRAPHS * DOUT), TB, 0, stream>>>(sums, NGRAPHS * DOUT);
  zero_f32<<<blocks(NGRAPHS), TB, 0, stream>>>(cnts, NGRAPHS);
  pool_accum<<<blocks((long long)NNODES * 32), TB, 0, stream>>>(a, bat, sums, cnts, NNODES);
  pool_final<<<blocks(NGRAPHS * DOUT), TB, 0, stream>>>(sums, cnts, out);
}